// WMSA_36730560315713
// MI455X (gfx1250) — compile-verified
//
#include <hip/hip_runtime.h>
#include <math.h>

typedef __attribute__((ext_vector_type(2))) float v2f;
typedef __attribute__((ext_vector_type(8))) float v8f;

#define BATCH 8
#define SEQ   16384
#define CDIM  256
#define P     64
#define NH    8
#define HD    32
#define NWIN  (SEQ / P)      /* 256 windows per batch */
#define SHIFT (P / 2)        /* 32 */

#define XST 260   /* x-window LDS row stride (floats): 260%64=4 -> conflict-free row gather */
#define QST 36    /* q/k/v & oh LDS row stride: 36r mod 64 distinct for r=0..15 */
#define SST 68    /* sim LDS row stride */

__global__ __launch_bounds__(256, 4)
void wmsa_fused(const float* __restrict__ x,
                const float* __restrict__ w_qkv,
                const float* __restrict__ b_qkv,
                const float* __restrict__ rel_pos,
                const float* __restrict__ w_out,
                const float* __restrict__ b_out,
                float* __restrict__ out)
{
    __shared__ float xs[P * XST];          /* 66560 B : rolled x window 64x256 */
    __shared__ float qkvb[3 * P * QST];    /* 27648 B : q,k,v for current head 64x32 each */
    __shared__ float simb[P * SST];        /* 17408 B : sim / probs 64x64 */
    __shared__ float oh[P * QST];          /*  9216 B : per-head attention output 64x32 */
    __shared__ float relsh[2 * P];         /*   512 B : rel_pos row (127 used) */
    /* total ~121 KB -> 2 workgroups per 320KB WGP */

    const int bb   = blockIdx.x / NWIN;
    const int win  = blockIdx.x % NWIN;
    const int tid  = threadIdx.x;
    const int wv   = tid >> 5;             /* wave id 0..7 (wave32) */
    const int lane = tid & 31;
    const int half = lane >> 4;            /* 0: lanes 0-15, 1: lanes 16-31 */
    const int idx  = lane & 15;
    const bool lastwin = (win == NWIN - 1);

    /* ---- stage 1: coalesced load of rolled x window into LDS ---- */
    const size_t xbase = (size_t)bb * SEQ * CDIM;
    for (int it = 0; it < 16; ++it) {
        int fi  = it * 256 + tid;          /* float4 index, 64 rows x 64 f4/row */
        int row = fi >> 6;
        int c4  = fi & 63;
        int tok = (win * P + row + SHIFT) & (SEQ - 1);   /* roll(-32) */
        float4 v = *(const float4*)(x + xbase + (size_t)tok * CDIM + c4 * 4);
        *(float4*)(&xs[row * XST + c4 * 4]) = v;
    }
    __syncthreads();

    const float scale = 0.17677669529663687f;  /* 32^-0.5 */

    /* persistent output-projection accumulators: wave's M-tile x 8 N-tiles */
    const int omt  = wv >> 1;
    const int ontb = (wv & 1) * 8;
    v8f pacc[8] = {};

    for (int h = 0; h < NH; ++h) {
        if (tid < 2 * P - 1) relsh[tid] = rel_pos[h * (2 * P - 1) + tid];

        /* ---- QKV projection: 24 tiles = 4 M-tiles x 6 (mat,nt); wave pair shares M-tile.
           Each wave: 3 accumulators sharing one A fragment per k-step. ---- */
        {
            const int mt = wv >> 1;
            const int js = (wv & 1) * 3;           /* j -> (mat=j>>1, nt=j&1) */
            int jrow[3];
            const float* brow[3];
            #pragma unroll
            for (int i = 0; i < 3; ++i) {
                int j  = js + i;
                jrow[i] = (j >> 1) * 256 + h * HD + (j & 1) * 16 + idx;
                brow[i] = w_qkv + (size_t)jrow[i] * CDIM + 2 * half;
            }
            const float* arow = &xs[(mt * 16 + idx) * XST + 2 * half];
            v8f acc[3] = {};
            for (int k0 = 0; k0 < CDIM; k0 += 4) {
                v2f a = *(const v2f*)(arow + k0);
                #pragma unroll
                for (int i = 0; i < 3; ++i) {
                    v2f b = *(const v2f*)(brow[i] + k0);
                    acc[i] = __builtin_amdgcn_wmma_f32_16x16x4_f32(
                                false, a, false, b, (short)0, acc[i], false, false);
                }
            }
            #pragma unroll
            for (int i = 0; i < 3; ++i) {
                int j      = js + i;
                float bias = b_qkv[jrow[i]];
                int n      = (j & 1) * 16 + idx;
                float* dst = &qkvb[(j >> 1) * (P * QST)];
                #pragma unroll
                for (int r = 0; r < 8; ++r) {
                    int m = mt * 16 + r + 8 * half;
                    dst[m * QST + n] = acc[i][r] + bias;
                }
            }
        }
        __syncthreads();

        /* ---- sim = (q @ k^T)*scale + rel_pos, masked: 16 tiles, K=32.
           Wave: one M-tile, two N-tiles sharing the A fragment. ---- */
        {
            const int mt  = wv >> 1;
            const int ntb = (wv & 1) * 2;
            const float* arow  = &qkvb[0 * P * QST + (mt * 16 + idx) * QST + 2 * half];
            const float* brow0 = &qkvb[1 * P * QST + ((ntb + 0) * 16 + idx) * QST + 2 * half];
            const float* brow1 = &qkvb[1 * P * QST + ((ntb + 1) * 16 + idx) * QST + 2 * half];
            v8f acc0 = {}, acc1 = {};
            for (int k0 = 0; k0 < HD; k0 += 4) {
                v2f a  = *(const v2f*)(arow + k0);
                v2f b0 = *(const v2f*)(brow0 + k0);
                v2f b1 = *(const v2f*)(brow1 + k0);
                acc0 = __builtin_amdgcn_wmma_f32_16x16x4_f32(
                            false, a, false, b0, (short)0, acc0, false, false);
                acc1 = __builtin_amdgcn_wmma_f32_16x16x4_f32(
                            false, a, false, b1, (short)0, acc1, false, false);
            }
            #pragma unroll
            for (int r = 0; r < 8; ++r) {
                int m  = mt * 16 + r + 8 * half;
                int n0 = (ntb + 0) * 16 + idx;
                int n1 = (ntb + 1) * 16 + idx;
                float v0 = acc0[r] * scale + relsh[m - n0 + P - 1];
                float v1 = acc1[r] * scale + relsh[m - n1 + P - 1];
                if (lastwin) {
                    if ((m < SHIFT) != (n0 < SHIFT)) v0 = -INFINITY;
                    if ((m < SHIFT) != (n1 < SHIFT)) v1 = -INFINITY;
                }
                simb[m * SST + n0] = v0;
                simb[m * SST + n1] = v1;
            }
        }
        __syncthreads();

        /* ---- softmax: 4 lanes per row (same wave), shfl_xor reduction ---- */
        {
            int row = tid >> 2;
            float* rp = &simb[row * SST + (tid & 3) * 16];
            float mx = -INFINITY;
            #pragma unroll
            for (int j = 0; j < 16; ++j) mx = fmaxf(mx, rp[j]);
            mx = fmaxf(mx, __shfl_xor(mx, 1, 32));
            mx = fmaxf(mx, __shfl_xor(mx, 2, 32));
            float s = 0.0f;
            #pragma unroll
            for (int j = 0; j < 16; ++j) { float e = __expf(rp[j] - mx); rp[j] = e; s += e; }
            s += __shfl_xor(s, 1, 32);
            s += __shfl_xor(s, 2, 32);
            float inv = 1.0f / s;
            #pragma unroll
            for (int j = 0; j < 16; ++j) rp[j] *= inv;
        }
        __syncthreads();

        /* ---- oh = probs @ v: 8 tiles (4 M x 2 N), K=64; one tile per wave ---- */
        {
            int mt = wv >> 1;
            int nt = wv & 1;
            int n  = nt * 16 + idx;
            const float* arow = &simb[(mt * 16 + idx) * SST + 2 * half];
            const float* vb   = &qkvb[2 * P * QST];
            v8f acc = {};
            for (int k0 = 0; k0 < P; k0 += 4) {
                v2f a = *(const v2f*)(arow + k0);
                v2f b;
                b.x = vb[(k0 + 2 * half)     * QST + n];
                b.y = vb[(k0 + 2 * half + 1) * QST + n];
                acc = __builtin_amdgcn_wmma_f32_16x16x4_f32(
                        false, a, false, b, (short)0, acc, false, false);
            }
            #pragma unroll
            for (int r = 0; r < 8; ++r) {
                int m = mt * 16 + r + 8 * half;
                oh[m * QST + n] = acc[r];
            }
        }
        __syncthreads();

        /* ---- partial output projection: pacc += oh @ w_out[:, h*32:(h+1)*32].T
           (K=32 slice of the full K=256 reduction). Wave: one M-tile, 8 N-tiles
           sharing one A fragment; B N-tile offsets are compile-time immediates.
           No trailing barrier needed: oh isn't overwritten until after the next
           head's three barriers. ---- */
        {
            const float* arow  = &oh[(omt * 16 + idx) * QST + 2 * half];
            const float* bbase = w_out + (size_t)(ontb * 16 + idx) * CDIM + h * HD + 2 * half;
            for (int k0 = 0; k0 < HD; k0 += 4) {
                v2f a = *(const v2f*)(arow + k0);
                #pragma unroll
                for (int i = 0; i < 8; ++i) {
                    v2f b = *(const v2f*)(bbase + (size_t)i * 16 * CDIM + k0);
                    pacc[i] = __builtin_amdgcn_wmma_f32_16x16x4_f32(
                                false, a, false, b, (short)0, pacc[i], false, false);
                }
            }
        }
    }

    /* ---- epilogue: add bias, store with roll(+32) folded into token index ---- */
    #pragma unroll
    for (int i = 0; i < 8; ++i) {
        int n = (ontb + i) * 16 + idx;
        float bias = b_out[n];
        #pragma unroll
        for (int r = 0; r < 8; ++r) {
            int m   = omt * 16 + r + 8 * half;
            int tok = (win * P + m + SHIFT) & (SEQ - 1);
            out[((size_t)bb * SEQ + tok) * CDIM + n] = pacc[i][r] + bias;
        }
    }
}

extern "C" void kernel_launch(void* const* d_in, const int* in_sizes, int n_in,
                              void* d_out, int out_size, void* d_ws, size_t ws_size,
                              hipStream_t stream) {
    const float* x       = (const float*)d_in[0];
    const float* w_qkv   = (const float*)d_in[1];
    const float* b_qkv   = (const float*)d_in[2];
    const float* rel_pos = (const float*)d_in[3];
    const float* w_out   = (const float*)d_in[4];
    const float* b_out   = (const float*)d_in[5];
    float* out = (float*)d_out;
    (void)in_sizes; (void)n_in; (void)out_size; (void)d_ws; (void)ws_size;

    dim3 grid(BATCH * NWIN);   /* 2048 workgroups: one per (batch, window) */
    dim3 block(256);           /* 8 wave32 waves */
    wmsa_fused<<<grid, block, 0, stream>>>(x, w_qkv, b_qkv, rel_pos, w_out, b_out, out);
}